// CustomModel_1159641170247
// MI455X (gfx1250) — compile-verified
//
#include <hip/hip_runtime.h>

// CDNA5 / gfx1250, wave32. D = A x B + C via V_WMMA_F32_16X16X4_F32 (exact f32).
// ~2 KB in / 200 B out, ~2 KFLOP: launch-latency bound on a 23.3 TB/s part.
// -> one workgroup (4 waves), cooperative LDS stage (each distinct transcendental
//    computed once), one barrier, then WMMA + ds-fed epilogue.
typedef __attribute__((ext_vector_type(2))) float v2f;
typedef __attribute__((ext_vector_type(8))) float v8f;

#define KB_EV    8.62e-05f
#define EPS0DOT  1.0e4f
#define PARAM_M  2.733f

// GROUP_COUNTS = [1,2,8,7,6,9,17]: group of row m via 6 compares (no memory).
__device__ __forceinline__ int groupOf(int m) {
    return (m >= 1) + (m >= 3) + (m >= 11) + (m >= 18) + (m >= 24) + (m >= 33);
}

__global__ __launch_bounds__(128) void yield_stress_wmma(
    const float* __restrict__ lsr,      // (50,6) row-major
    const float* __restrict__ temp,     // (50,)
    const float* __restrict__ srate,    // (50,)
    const float* __restrict__ grain,    // (50,)
    const float* __restrict__ w21,      // (21,)
    const float* __restrict__ raw_dH,   // (7,6)
    const float* __restrict__ raw_khp,  // (7,)
    float* __restrict__ out)            // (50,)
{
    __shared__ float sW[36];    // W(k,n) = 0.1 + exp(w21[triu(k,n)])
    __shared__ float sDH[42];   // (0.1 + 4.9*sigmoid(raw_dH))^(-2/3)
    __shared__ float sPCM[50];  // (KB*T*ln(eps0/srate))^(2/3) per row
    __shared__ float sHP[50];   // exp(khp[g(m)]) * grain^-1/2 per row

    const int tid = threadIdx.x;

    // ---- cooperative stage: each distinct value computed exactly once ----
    if (tid < 50) {                               // per-row scalars
        const float cm = KB_EV * temp[tid] * __logf(EPS0DOT / srate[tid]);
        sPCM[tid] = __powf(cm, 0.66666667f);
        sHP[tid]  = __expf(raw_khp[groupOf(tid)]) * rsqrtf(grain[tid]);
    }
    if (tid >= 64 && tid < 112) {                 // 36 W entries on 48 slots
        const unsigned u = tid - 64, k = u >> 3, n = u & 7;
        if (n < 6) {
            const unsigned a = k < n ? k : n;
            const unsigned b = k < n ? n : k;
            const unsigned idx = a * 6u - ((a * (a - 1u)) >> 1) + (b - a);
            sW[k * 6 + n] = 0.1f + __expf(w21[idx]);
        }
    }
    if (tid >= 72) {                              // 42 dH entries on 56 slots
        const unsigned u = tid - 72, g = u >> 3, n = u & 7;
        if (n < 6) {
            const float sig = 1.0f / (1.0f + __expf(-raw_dH[g * 6 + n]));
            sDH[g * 6 + n] = __powf(0.1f + 4.9f * sig, -0.66666667f);
        }
    }
    __syncthreads();                              // EXEC all-ones again for WMMA

    const int wave = tid >> 5;                    // 4 waves: one 16-row M tile
    const int lane = tid & 31;
    const int half = lane >> 4;                   // K pair (A/B), M+8 (C/D)
    const int l15  = lane & 15;
    const int rowBase = wave * 16;

    const int   ncl   = l15 < 6 ? l15 : 5;        // this lane's N column
    const float nmask = (l15 < 6) ? 1.0f : 0.0f;

    // ---- A/B fragments + 2x WMMA over K=0..7 (K,N >= 6 zero-padded) ----
    // f32 A 16x4: lanes 0-15 hold K={0,1}, lanes 16-31 K={2,3}; B mirrors it.
    const int   mrow  = rowBase + l15;
    const int   mc    = mrow < 50 ? mrow : 49;
    const float mmask = mrow < 50 ? 1.0f : 0.0f;
    const float* lrow = lsr + mc * 6;             // 24B row stride

    const int kb = half * 2;                      // slice 0: K = kb, kb+1 (<6)
    v2f a0 = *(const v2f*)(lrow + kb);            // 8B-aligned (24m + {0,8})
    a0.x *= mmask;  a0.y *= mmask;
    v2f b0;
    b0.x = nmask * sW[ kb      * 6 + ncl];
    b0.y = nmask * sW[(kb + 1) * 6 + ncl];

    v8f c = {};
    c = __builtin_amdgcn_wmma_f32_16x16x4_f32(false, a0, false, b0,
                                              (short)0, c, false, false);

    // slice 1: K = 4,5 in low half; high half (K=6,7) fully masked to zero
    const float kmask = half ? 0.0f : 1.0f;
    const float am = mmask * kmask, bm = nmask * kmask;
    v2f a1 = *(const v2f*)(lrow + 4);             // 8B-aligned (24m + 16)
    a1.x *= am;  a1.y *= am;
    v2f b1;
    b1.x = bm * sW[24 + ncl];                     // W(4, n)
    b1.y = bm * sW[30 + ncl];                     // W(5, n)
    c = __builtin_amdgcn_wmma_f32_16x16x4_f32(false, a1, false, b1,
                                              (short)0, c, false, false);

    // ---- thermal scaling per C element, 16-lane xor-reduce over N ----
    // C/D layout: VGPR r holds row (rowBase + half*8 + r), column N = l15.
    float tau[8];
#pragma unroll
    for (int r = 0; r < 8; ++r) {
        const int m   = rowBase + half * 8 + r;
        const int mcl = m < 50 ? m : 49;          // padded rows: c[r]==0 anyway
        const int g   = groupOf(mcl);
        const float d = sDH[g * 6 + ncl];
        const float p = sPCM[mcl];
        float v = c[r] * (1.0f - p * d) * nmask;
#pragma unroll
        for (int off = 1; off < 16; off <<= 1)    // N>=6 terms are zero
            v += __shfl_xor(v, off, 16);
        tau[r] = v;
    }

    // ---- writeback: lanes l15 = 0..7 of each half, one row each ----
    float myTau = tau[0];
#pragma unroll
    for (int r = 1; r < 8; ++r)
        myTau = ((l15 & 7) == r) ? tau[r] : myTau;

    const int rlane = rowBase + half * 8 + (l15 & 7);
    if (l15 < 8 && rlane < 50)
        out[rlane] = myTau * PARAM_M + sHP[rlane];
}

extern "C" void kernel_launch(void* const* d_in, const int* in_sizes, int n_in,
                              void* d_out, int out_size, void* d_ws, size_t ws_size,
                              hipStream_t stream) {
    const float* lsr   = (const float*)d_in[0];  // (50,6)
    const float* temp  = (const float*)d_in[1];  // (50,)
    const float* srate = (const float*)d_in[2];  // (50,)
    const float* grain = (const float*)d_in[3];  // (50,)
    const float* w21   = (const float*)d_in[4];  // (21,)
    const float* dH    = (const float*)d_in[5];  // (7,6)
    const float* khp   = (const float*)d_in[6];  // (7,)
    yield_stress_wmma<<<1, 128, 0, stream>>>(lsr, temp, srate, grain, w21, dH, khp,
                                             (float*)d_out);
}